// TaskAlignedAssigner_14826227106667
// MI455X (gfx1250) — compile-verified
//
#include <hip/hip_runtime.h>
#include <hip/hip_bf16.h>
#include <math.h>

// Problem constants (from reference)
#define B_      8
#define N_      65536
#define M_      64
#define C_      2
#define TOPK_   13
#define EPS_    1e-9f
// -ln(MIN_IOU) with MIN_IOU = 0.05
#define NEG_LN_MIN_IOU_ 2.9957322735539909f

#define CHUNK_  2048
#define NCHUNK_ (N_ / CHUNK_)   // 32 chunks per batch
#define NSUB_   (CHUNK_ / 128)  // 16 subtiles of 128 preds

typedef __attribute__((ext_vector_type(2))) float v2f;
typedef __attribute__((ext_vector_type(8))) float v8f;

// Maintain a descending sorted top-13 in registers with a single unrolled
// bubble pass per insert (static indices only -> no scratch spills).
__device__ __forceinline__ void insert13(float tk[TOPK_], float v) {
  if (v <= tk[TOPK_ - 1]) return;
  tk[TOPK_ - 1] = v;
#pragma unroll
  for (int i = TOPK_ - 1; i > 0; --i) {
    float a = tk[i - 1], c = tk[i];
    tk[i - 1] = fmaxf(a, c);
    tk[i]     = fminf(a, c);
  }
}

// Async global->LDS copy, 16B per lane (CDNA5 GLOBAL_LOAD_ASYNC_TO_LDS_B128,
// GVS addressing: mem = SADDR + VADDR + IOFFSET). Tracked by ASYNCcnt.
// lds_dst generic pointer: low 32 bits are the LDS byte offset (ISA 10.2).
__device__ __forceinline__ void async_copy_b128(void* lds_dst, const void* gbase,
                                                unsigned voff_bytes) {
  unsigned ldsoff = (unsigned)(uintptr_t)lds_dst;
  asm volatile("global_load_async_to_lds_b128 %0, %1, %2"
               :: "v"(ldsoff), "v"(voff_bytes), "s"(gbase)
               : "memory");
}

// ---------------------------------------------------------------------------
// K0: zero atomic-max accumulators (ws is poisoned by the harness each run)
// ---------------------------------------------------------------------------
__global__ void k0_init(unsigned* __restrict__ maxmet,
                        unsigned* __restrict__ maxiou) {
  int t = blockIdx.x * 256 + threadIdx.x;
  if (t < B_ * M_) { maxmet[t] = 0u; maxiou[t] = 0u; }
}

// ---------------------------------------------------------------------------
// K1: WMMA metric field + per-chunk per-gt top-13 partials
//   grid = (NCHUNK_, B_), block = 256 (8 wave32)
//   One V_WMMA_F32_16X16X4_F32 computes, for a 16x16 (pred x gt) tile,
//   ||t||^2 - 2 p.t  via  A = (x,y,z,1), B = (-2tx,-2ty,-2tz,||t||^2).
//   Raw pred/anchor/score bytes double-buffered via async global->LDS DMA.
// ---------------------------------------------------------------------------
__global__ __launch_bounds__(256)
void k1_metrics_topk(const float* __restrict__ pred_scores,
                     const float* __restrict__ pred_centers,
                     const float* __restrict__ anchor_points,
                     const int*   __restrict__ true_labels,
                     const float* __restrict__ true_centers,
                     const float* __restrict__ true_sigmas,
                     float* __restrict__ partials) {
  __shared__ float  s_bm[M_][4];     // B-fragment source: {-2x,-2y,-2z,|t|^2}
  __shared__ float  s_kc[M_];        // 1/(2 sigma^2)
  __shared__ float  s_ath[M_];       // 2 sigma^2 * (-ln MIN_IOU): in-gts dist threshold
  __shared__ int    s_lab[M_];
  __shared__ float4 s_pinfo[128];    // {|p|^2, |a|^2, score0, score1}
  __shared__ float  s_px[128], s_py[128], s_pz[128];
  __shared__ float  s_ax[128], s_ay[128], s_az[128];
  __shared__ float  s_mblk[M_ * 128];  // metric block [gt][pred]; reused for merge
  // double-buffered raw staging filled by async DMA (4096 B per buffer)
  __shared__ float  s_rawpc[2][128 * 3];  // 96 lanes x 16B
  __shared__ float  s_rawax[2][128 * 3];  // 96 lanes x 16B
  __shared__ float  s_rawsc[2][128 * 2];  // 64 lanes x 16B

  const int b     = blockIdx.y;
  const int chunk = blockIdx.x;
  const int base  = chunk * CHUNK_;
  const int t     = threadIdx.x;
  const int lane  = t & 31;
  const int w     = t >> 5;          // wave id 0..7
  const bool hi   = (lane >= 16);
  const int l16   = lane & 15;

  // Issue one async b128 per lane: waves 0-2 copy pred_centers (1536 B),
  // waves 3-5 copy anchors (1536 B), waves 6-7 copy scores (1024 B).
  // SADDR base is wave-uniform; VADDR carries the per-lane byte offset.
  auto issue_copy = [&](int sub, int buf) {
    const int p0 = base + sub * 128;
    if (t < 96) {
      const float* g = pred_centers + ((size_t)b * N_ + p0) * 3;
      async_copy_b128(&s_rawpc[buf][t * 4], g, (unsigned)(t * 16));
    } else if (t < 192) {
      const float* g = anchor_points + (size_t)p0 * 3;
      async_copy_b128(&s_rawax[buf][(t - 96) * 4], g, (unsigned)((t - 96) * 16));
    } else {
      const float* g = pred_scores + ((size_t)b * N_ + p0) * C_;
      async_copy_b128(&s_rawsc[buf][(t - 192) * 4], g, (unsigned)((t - 192) * 16));
    }
  };

  issue_copy(0, 0);  // prefetch first subtile

  if (t < M_) {
    float tx = true_centers[(b * M_ + t) * 3 + 0];
    float ty = true_centers[(b * M_ + t) * 3 + 1];
    float tz = true_centers[(b * M_ + t) * 3 + 2];
    s_bm[t][0] = -2.f * tx;
    s_bm[t][1] = -2.f * ty;
    s_bm[t][2] = -2.f * tz;
    s_bm[t][3] = tx * tx + ty * ty + tz * tz;
    float sg = true_sigmas[b * M_ + t];
    s_kc[t]  = 1.f / (2.f * sg * sg);
    s_ath[t] = 2.f * sg * sg * NEG_LN_MIN_IOU_;   // d^2 < this  <=>  oks > MIN_IOU
    s_lab[t] = true_labels[b * M_ + t];
  }

  float tk[TOPK_];
#pragma unroll
  for (int i = 0; i < TOPK_; ++i) tk[i] = -1e30f;
  const int mrow  = t & (M_ - 1);    // gt row this thread scans
  const int quart = t >> 6;          // which 32-pred quarter

  for (int sub = 0; sub < NSUB_; ++sub) {
    const int cur = sub & 1;
    // Prefetch next subtile into the other buffer, then wait for the older
    // (current) copy: async loads complete in order, so asynccnt<=1 means
    // the current buffer's DMA is done.
    if (sub + 1 < NSUB_) {
      issue_copy(sub + 1, cur ^ 1);
      asm volatile("s_wait_asynccnt 0x1" ::: "memory");
    } else {
      asm volatile("s_wait_asynccnt 0x0" ::: "memory");
    }
    __syncthreads();   // all waves' DMA for `cur` now visible in LDS

    if (t < 128) {
      float x = s_rawpc[cur][t * 3 + 0];
      float y = s_rawpc[cur][t * 3 + 1];
      float z = s_rawpc[cur][t * 3 + 2];
      s_px[t] = x; s_py[t] = y; s_pz[t] = z;
      float ax = s_rawax[cur][t * 3 + 0];
      float ay = s_rawax[cur][t * 3 + 1];
      float az = s_rawax[cur][t * 3 + 2];
      s_ax[t] = ax; s_ay[t] = ay; s_az[t] = az;
      float4 pi4;
      pi4.x = x * x + y * y + z * z;
      pi4.y = ax * ax + ay * ay + az * az;
      pi4.z = s_rawsc[cur][t * 2 + 0];
      pi4.w = s_rawsc[cur][t * 2 + 1];
      s_pinfo[t] = pi4;
    }
    __syncthreads();

    // A fragment (16x4 f32): lanes 0-15 hold K=0,1 = (x,y); lanes 16-31 K=2,3 = (z,1)
    const int pidx = (w << 4) + l16;
    v2f Ap, Aa;
    Ap.x = hi ? s_pz[pidx] : s_px[pidx];
    Ap.y = hi ? 1.0f       : s_py[pidx];
    Aa.x = hi ? s_az[pidx] : s_ax[pidx];
    Aa.y = hi ? 1.0f       : s_ay[pidx];

#pragma unroll
    for (int g = 0; g < M_ / 16; ++g) {
      const int gi = (g << 4) + l16;
      // B fragment (4x16 f32): lanes 0-15 rows K=0,1; lanes 16-31 rows K=2,3
      v2f Bm;
      Bm.x = hi ? s_bm[gi][2] : s_bm[gi][0];
      Bm.y = hi ? s_bm[gi][3] : s_bm[gi][1];
      v8f c  = {0.f, 0.f, 0.f, 0.f, 0.f, 0.f, 0.f, 0.f};
      v8f ca = {0.f, 0.f, 0.f, 0.f, 0.f, 0.f, 0.f, 0.f};
      // D = A x B + C : yields ||t||^2 - 2 p.t per (pred,gt) element
      c  = __builtin_amdgcn_wmma_f32_16x16x4_f32(false, Ap, false, Bm, (short)0, c,  false, false);
      ca = __builtin_amdgcn_wmma_f32_16x16x4_f32(false, Aa, false, Bm, (short)0, ca, false, false);

      const float kc  = s_kc[gi];
      const float ath = s_ath[gi];
      const int   lg  = s_lab[gi];
#pragma unroll
      for (int r = 0; r < 8; ++r) {
        // C layout: vgpr r / lane half -> pred (r + 8*hi), gt (lane&15)
        const int pit = r + (hi ? 8 : 0);
        float4 pf = s_pinfo[(w << 4) + pit];
        float d2  = pf.x + c[r];    // ||p - t||^2
        float da  = pf.y + ca[r];   // ||a - t||^2
        float iou = __expf(-d2 * kc);
        float i2  = iou * iou;
        float i6  = i2 * i2 * i2;   // iou^BETA, BETA=6
        float sc  = lg ? pf.w : pf.z;
        float metr = (da < ath) ? (sc * i6) : 0.0f;   // alignment * is_in_gts
        s_mblk[gi * 128 + (w << 4) + pit] = metr;
      }
    }
    __syncthreads();

    // per-thread running top-13 over this 64x128 block (4 threads per gt row)
    {
      const float* row = &s_mblk[mrow * 128 + quart * 32];
#pragma unroll 4
      for (int i = 0; i < 32; ++i) insert13(tk, row[i]);
    }
    __syncthreads();
  }

  // Merge the 4 per-quarter sorted lists per gt -> chunk top-13 partial.
  float* s_lists = s_mblk;  // reuse (256*13 = 3328 <= 8192 floats)
#pragma unroll
  for (int i = 0; i < TOPK_; ++i) s_lists[t * TOPK_ + i] = tk[i];
  __syncthreads();
  if (t < M_) {
    const float* L0 = &s_lists[(t       ) * TOPK_];
    const float* L1 = &s_lists[(t +  64 ) * TOPK_];
    const float* L2 = &s_lists[(t + 128 ) * TOPK_];
    const float* L3 = &s_lists[(t + 192 ) * TOPK_];
    int i0 = 0, i1 = 0, i2 = 0, i3 = 0;
    const size_t off = (((size_t)b * NCHUNK_ + chunk) * M_ + t) * TOPK_;
    for (int k = 0; k < TOPK_; ++k) {
      float v0 = (i0 < TOPK_) ? L0[i0] : -1e30f;
      float v1 = (i1 < TOPK_) ? L1[i1] : -1e30f;
      float v2 = (i2 < TOPK_) ? L2[i2] : -1e30f;
      float v3 = (i3 < TOPK_) ? L3[i3] : -1e30f;
      float best = v0; int bj = 0;
      if (v1 > best) { best = v1; bj = 1; }
      if (v2 > best) { best = v2; bj = 2; }
      if (v3 > best) { best = v3; bj = 3; }
      if      (bj == 0) ++i0;
      else if (bj == 1) ++i1;
      else if (bj == 2) ++i2;
      else              ++i3;
      partials[off + k] = best;
    }
  }
}

// ---------------------------------------------------------------------------
// K2: reduce chunk partials -> 13th-largest metric threshold per (b, m)
// ---------------------------------------------------------------------------
__global__ __launch_bounds__(256)
void k2_thresh(const float* __restrict__ partials, float* __restrict__ thresh) {
  int tid = blockIdx.x * 256 + threadIdx.x;
  if (tid >= B_ * M_) return;
  const int b = tid >> 6;
  const int m = tid & (M_ - 1);
  float tk[TOPK_];
#pragma unroll
  for (int i = 0; i < TOPK_; ++i) tk[i] = -1e30f;
  for (int c = 0; c < NCHUNK_; ++c) {
    const size_t off = (((size_t)b * NCHUNK_ + c) * M_ + m) * TOPK_;
#pragma unroll
    for (int k = 0; k < TOPK_; ++k) insert13(tk, partials[off + k]);
  }
  thresh[tid] = tk[TOPK_ - 1];
}

// ---------------------------------------------------------------------------
// K3: per-anchor column resolution over all 64 gts
//   grid = (N_/256, B_), block = 256; one thread per anchor n
// ---------------------------------------------------------------------------
__global__ __launch_bounds__(256)
void k3_resolve(const float* __restrict__ pred_scores,
                const float* __restrict__ pred_centers,
                const float* __restrict__ anchor_points,
                const int*   __restrict__ true_labels,
                const float* __restrict__ true_centers,
                const float* __restrict__ true_sigmas,
                const float* __restrict__ pad_gt_mask,
                const int*   __restrict__ bg_index,
                const float* __restrict__ thresh,
                float* __restrict__ out_labels,
                float* __restrict__ out_points,
                float* __restrict__ out_sigmas,
                int*   __restrict__ asg_ws,
                float* __restrict__ amv_ws,
                float* __restrict__ iov_ws,
                unsigned* __restrict__ maxmet,
                unsigned* __restrict__ maxiou) {
  __shared__ float s_tx[M_], s_ty[M_], s_tz[M_];
  __shared__ float s_kc[M_], s_ath[M_], s_sig[M_], s_thr[M_], s_pad[M_];
  __shared__ int   s_lab[M_];

  const int b = blockIdx.y;
  const int t = threadIdx.x;
  if (t < M_) {
    s_tx[t]  = true_centers[(b * M_ + t) * 3 + 0];
    s_ty[t]  = true_centers[(b * M_ + t) * 3 + 1];
    s_tz[t]  = true_centers[(b * M_ + t) * 3 + 2];
    float sg = true_sigmas[b * M_ + t];
    s_sig[t] = sg;
    s_kc[t]  = 1.f / (2.f * sg * sg);
    s_ath[t] = 2.f * sg * sg * NEG_LN_MIN_IOU_;
    s_thr[t] = thresh[b * M_ + t];
    s_pad[t] = pad_gt_mask[b * M_ + t];
    s_lab[t] = true_labels[b * M_ + t];
  }
  __syncthreads();

  const size_t n  = (size_t)blockIdx.x * 256 + t;
  const size_t bn = (size_t)b * N_ + n;
  const float px = pred_centers[bn * 3 + 0];
  const float py = pred_centers[bn * 3 + 1];
  const float pz = pred_centers[bn * 3 + 2];
  const float ax = anchor_points[n * 3 + 0];
  const float ay = anchor_points[n * 3 + 1];
  const float az = anchor_points[n * 3 + 2];
  const float s0 = pred_scores[bn * C_ + 0];
  const float s1 = pred_scores[bn * C_ + 1];

  int   cnt = 0, firstm = -1, bim = 0;
  float fmet = 0.f, fiou = 0.f, bmet = 0.f, bestiou = -1e30f;

  for (int m = 0; m < M_; ++m) {
    float dx = px - s_tx[m], dy = py - s_ty[m], dz = pz - s_tz[m];
    float d2 = dx * dx + dy * dy + dz * dz;
    float iou = __expf(-d2 * s_kc[m]);
    float adx = ax - s_tx[m], ady = ay - s_ty[m], adz = az - s_tz[m];
    float da2 = adx * adx + ady * ady + adz * adz;
    float i2 = iou * iou;
    float i6 = i2 * i2 * i2;
    float met = (s_lab[m] ? s1 : s0) * i6;       // raw alignment metric
    bool  ing = (da2 < s_ath[m]);
    float metg = ing ? met : 0.f;
    // top-k as threshold; metg>0 guard avoids degenerate all-pass on zero rows
    bool mask = ing && (metg >= s_thr[m]) && (metg > 0.f) && (s_pad[m] > 0.5f);
    if (mask) { if (firstm < 0) { firstm = m; fmet = met; fiou = iou; } ++cnt; }
    if (iou > bestiou) { bestiou = iou; bim = m; bmet = met; }  // first-max wins
  }

  int asg; float amv, iov; bool pos;
  if (cnt > 1)       { asg = bim;    amv = bmet; iov = bestiou; pos = true;  }
  else if (cnt == 1) { asg = firstm; amv = fmet; iov = fiou;    pos = true;  }
  else               { asg = 0;      amv = 0.f;  iov = 0.f;     pos = false; }

  out_labels[bn] = pos ? (float)s_lab[asg] : (float)(*bg_index);
  out_points[bn * 3 + 0] = s_tx[asg];   // argmax of zeros -> gt 0, like reference
  out_points[bn * 3 + 1] = s_ty[asg];
  out_points[bn * 3 + 2] = s_tz[asg];
  out_sigmas[bn] = s_sig[asg];
  asg_ws[bn] = pos ? asg : -1;
  amv_ws[bn] = amv;
  iov_ws[bn] = iov;
  if (pos) {
    atomicMax(&maxmet[b * M_ + asg], __float_as_uint(amv));  // vals >= 0: uint order ok
    atomicMax(&maxiou[b * M_ + asg], __float_as_uint(iov));
  }
}

// ---------------------------------------------------------------------------
// K4: assigned_scores = one_hot(label)[keep] * am/(max_metric+eps)*max_iou
// ---------------------------------------------------------------------------
__global__ __launch_bounds__(256)
void k4_scores(const int* __restrict__ true_labels,
               const int* __restrict__ asg_ws,
               const float* __restrict__ amv_ws,
               const float* __restrict__ iov_ws,
               const unsigned* __restrict__ maxmet,
               const unsigned* __restrict__ maxiou,
               float* __restrict__ out_scores) {
  const size_t tid = (size_t)blockIdx.x * 256 + threadIdx.x;  // over B*N
  const int b = (int)(tid / N_);
  const int asg = asg_ws[tid];
  float sc0 = 0.f, sc1 = 0.f;
  if (asg >= 0) {
    float mm = __uint_as_float(maxmet[b * M_ + asg]);
    float mi = __uint_as_float(maxiou[b * M_ + asg]);
    float scale = amv_ws[tid] / (mm + EPS_) * mi;
    int lb = true_labels[b * M_ + asg];
    sc0 = (lb == 0) ? scale : 0.f;
    sc1 = (lb == 1) ? scale : 0.f;
  }
  out_scores[tid * C_ + 0] = sc0;
  out_scores[tid * C_ + 1] = sc1;
}

// ---------------------------------------------------------------------------
extern "C" void kernel_launch(void* const* d_in, const int* in_sizes, int n_in,
                              void* d_out, int out_size, void* d_ws, size_t ws_size,
                              hipStream_t stream) {
  const float* pred_scores   = (const float*)d_in[0];
  const float* pred_centers  = (const float*)d_in[1];
  const float* anchor_points = (const float*)d_in[2];
  const int*   true_labels   = (const int*)d_in[3];
  const float* true_centers  = (const float*)d_in[4];
  const float* true_sigmas   = (const float*)d_in[5];
  const float* pad_gt_mask   = (const float*)d_in[6];
  const int*   bg_index      = (const int*)d_in[7];

  // Output tuple concatenated flat (all as float):
  // labels (B,N) | points (B,N,3) | scores (B,N,2) | sigmas (B,N)
  float* out        = (float*)d_out;
  float* out_labels = out;
  float* out_points = out + (size_t)B_ * N_;
  float* out_scores = out + (size_t)B_ * N_ * 4;
  float* out_sigmas = out + (size_t)B_ * N_ * 6;

  // Workspace carve-up (~7.2 MB total)
  float*    partials = (float*)d_ws;                                   // B*NCHUNK*M*13
  float*    thresh   = partials + (size_t)B_ * NCHUNK_ * M_ * TOPK_;   // B*M
  unsigned* maxmet   = (unsigned*)(thresh + B_ * M_);                  // B*M
  unsigned* maxiou   = maxmet + B_ * M_;                               // B*M
  int*      asg_ws   = (int*)(maxiou + B_ * M_);                       // B*N
  float*    amv_ws   = (float*)(asg_ws + (size_t)B_ * N_);             // B*N
  float*    iov_ws   = amv_ws + (size_t)B_ * N_;                       // B*N

  k0_init<<<2, 256, 0, stream>>>(maxmet, maxiou);
  k1_metrics_topk<<<dim3(NCHUNK_, B_), 256, 0, stream>>>(
      pred_scores, pred_centers, anchor_points, true_labels, true_centers,
      true_sigmas, partials);
  k2_thresh<<<2, 256, 0, stream>>>(partials, thresh);
  k3_resolve<<<dim3(N_ / 256, B_), 256, 0, stream>>>(
      pred_scores, pred_centers, anchor_points, true_labels, true_centers,
      true_sigmas, pad_gt_mask, bg_index, thresh,
      out_labels, out_points, out_sigmas, asg_ws, amv_ws, iov_ws,
      maxmet, maxiou);
  k4_scores<<<(B_ * N_) / 256, 256, 0, stream>>>(
      true_labels, asg_ws, amv_ws, iov_ws, maxmet, maxiou, out_scores);
}